// GraphModel_11785390260437
// MI455X (gfx1250) — compile-verified
//
#include <hip/hip_runtime.h>
#include <hip/hip_bf16.h>

#define N_NODES 100000      // == 16 * 6250  -> no row guards needed
#define N_EDGES 3200000
#define N_GRAPHS 4096       // == 16 * 256
#define IN_DIM 74
#define HID 64
#define MLP_HID 128

typedef float v2f __attribute__((ext_vector_type(2)));
typedef float v8f __attribute__((ext_vector_type(8)));

// Full-precision CDNA5 matrix op: D(16x16,f32) = A(16x4,f32) * B(4x16,f32) + C
__device__ __forceinline__ v8f wmma4(v2f a, v2f b, v8f c) {
    return __builtin_amdgcn_wmma_f32_16x16x4_f32(
        /*neg_a=*/false, a, /*neg_b=*/false, b,
        /*c_mod=*/(short)0, c, /*reuse_a=*/false, /*reuse_b=*/false);
}

__device__ __forceinline__ v2f ld2(const float* __restrict__ p) {
    return *reinterpret_cast<const v2f*>(p);   // 8B-aligned (even k0, even row stride)
}

// ---------------------------------------------------------------------------
// Core: acc[NT] += Xrow(16 x K) @ B(K x NT*16), K compile-time, fully unrolled.
// A layout (16x4 f32): lane rc=row, half h: VGPR0=K k0, VGPR1=K k0+1, k0=k+2h.
// B layout (4x16 f32): lane rc=col, half h: rows k+2h, k+2h+1.
// No EXEC manipulation in the main loop; single guarded tail iff K%4 != 0.
// ---------------------------------------------------------------------------
template <int K, int NT>
__device__ __forceinline__ void gemm_core(const float* __restrict__ Xrow,  // &X[row*K]
                                          const float* __restrict__ B,     // [K x NT*16]
                                          int half, int rc, v8f acc[NT]) {
    static_assert(K % 4 == 0 || K % 4 == 2, "tail handling assumes K%4 in {0,2}");
    constexpr int KMAIN = K & ~3;
    constexpr int NCOL  = NT * 16;
#pragma unroll
    for (int k = 0; k < KMAIN; k += 4) {
        const int k0 = k + half * 2;
        const v2f a = ld2(Xrow + k0);
#pragma unroll
        for (int t = 0; t < NT; ++t) {
            const int col = t * 16 + rc;
            v2f b;
            b.x = B[(size_t)k0 * NCOL + col];
            b.y = B[(size_t)(k0 + 1) * NCOL + col];
            acc[t] = wmma4(a, b, acc[t]);
        }
    }
    if constexpr (K % 4 == 2) {
        // tail step k=KMAIN: only lane-half 0 (k0=KMAIN, KMAIN+1) is in-bounds
        v2f a = {0.0f, 0.0f};
        if (half == 0) a = ld2(Xrow + KMAIN);
#pragma unroll
        for (int t = 0; t < NT; ++t) {
            const int col = t * 16 + rc;
            v2f b = {0.0f, 0.0f};
            if (half == 0) {
                b.x = B[(size_t)KMAIN * NCOL + col];
                b.y = B[(size_t)(KMAIN + 1) * NCOL + col];
            }
            acc[t] = wmma4(a, b, acc[t]);   // EXEC restored to all-ones before WMMA
        }
    }
}

// ---------------------------------------------------------------------------
// XW = X @ W   (X: [nrows x K], W: [K x 64]); one wave = 16 rows x 64 cols
// ---------------------------------------------------------------------------
template <int K>
__global__ void gemm64_kernel(const float* __restrict__ X,
                              const float* __restrict__ W,
                              float* __restrict__ XW,
                              int nrows) {
    const int wave = blockIdx.x * (blockDim.x >> 5) + (threadIdx.x >> 5);
    const int lane = threadIdx.x & 31;
    const int half = lane >> 4;
    const int rc   = lane & 15;
    const int rowBase = wave * 16;
    if (rowBase >= nrows) return;           // uniform per wave

    v8f acc[4] = {v8f{0}, v8f{0}, v8f{0}, v8f{0}};
    gemm_core<K, 4>(X + (size_t)(rowBase + rc) * K, W, half, rc, acc);

#pragma unroll
    for (int t = 0; t < 4; ++t)
#pragma unroll
        for (int r = 0; r < 8; ++r) {
            const int m = rowBase + r + 8 * half;           // C: VGPR r -> M=r / 8+r
            XW[(size_t)m * HID + t * 16 + rc] = acc[t][r];  // nrows % 16 == 0: no guard
        }
}

// ---------------------------------------------------------------------------
// OUT = relu(AGG + b) + relu(X @ R + rb)   (residual GEMM fused)
// ---------------------------------------------------------------------------
template <int K>
__global__ void combine_kernel(const float* __restrict__ X,
                               const float* __restrict__ R,
                               const float* __restrict__ rb,
                               const float* __restrict__ AGG,
                               const float* __restrict__ b,
                               float* __restrict__ OUT,
                               int nrows) {
    const int wave = blockIdx.x * (blockDim.x >> 5) + (threadIdx.x >> 5);
    const int lane = threadIdx.x & 31;
    const int half = lane >> 4;
    const int rc   = lane & 15;
    const int rowBase = wave * 16;
    if (rowBase >= nrows) return;

    v8f acc[4] = {v8f{0}, v8f{0}, v8f{0}, v8f{0}};
    gemm_core<K, 4>(X + (size_t)(rowBase + rc) * K, R, half, rc, acc);

#pragma unroll
    for (int t = 0; t < 4; ++t) {
        const int   col   = t * 16 + rc;
        const float bias  = b[col];
        const float rbias = rb[col];
#pragma unroll
        for (int r = 0; r < 8; ++r) {
            const int m  = rowBase + r + 8 * half;
            const float hv = fmaxf(AGG[(size_t)m * HID + col] + bias, 0.0f);
            const float rv = fmaxf(acc[t][r] + rbias, 0.0f);
            OUT[(size_t)m * HID + col] = hv + rv;
        }
    }
}

// ---------------------------------------------------------------------------
// Edge scatter: AGG[dst[e]] += XW[src[e]]  (float4 chunks, f32 atomics).
// Working set (~51MB xw+agg + 25MB edges) is L2-resident on the 192MB L2.
// ---------------------------------------------------------------------------
__global__ void scatter_edges_kernel(const int* __restrict__ src,
                                     const int* __restrict__ dst,
                                     const float* __restrict__ XW,
                                     float* __restrict__ AGG) {
    const long long tid = (long long)blockIdx.x * blockDim.x + threadIdx.x;
    const int e = (int)(tid >> 4);
    const int q = (int)(tid & 15);
    if (e >= N_EDGES) return;
    const int s = src[e];
    const int d = dst[e];
    const float4 v = *reinterpret_cast<const float4*>(XW + (size_t)s * HID + q * 4);
    float* p = AGG + (size_t)d * HID + q * 4;
    unsafeAtomicAdd(p + 0, v.x);
    unsafeAtomicAdd(p + 1, v.y);
    unsafeAtomicAdd(p + 2, v.z);
    unsafeAtomicAdd(p + 3, v.w);
}

// ---------------------------------------------------------------------------
// SumPool: G[graph_ids[n]] += X[n]
// ---------------------------------------------------------------------------
__global__ void pool_kernel(const float* __restrict__ X,
                            const int* __restrict__ gid,
                            float* __restrict__ G) {
    const long long tid = (long long)blockIdx.x * blockDim.x + threadIdx.x;
    const int n = (int)(tid >> 4);
    const int q = (int)(tid & 15);
    if (n >= N_NODES) return;
    const int g = gid[n];
    const float4 v = *reinterpret_cast<const float4*>(X + (size_t)n * HID + q * 4);
    float* p = G + (size_t)g * HID + q * 4;
    unsafeAtomicAdd(p + 0, v.x);
    unsafeAtomicAdd(p + 1, v.y);
    unsafeAtomicAdd(p + 2, v.z);
    unsafeAtomicAdd(p + 3, v.w);
}

// ---------------------------------------------------------------------------
// H = relu(G @ Wm1 + bm1)   (G: [4096 x 64], Wm1: [64 x 128]); 8 N-tiles
// ---------------------------------------------------------------------------
__global__ void mlp1_kernel(const float* __restrict__ G,
                            const float* __restrict__ Wm1,
                            const float* __restrict__ bm1,
                            float* __restrict__ H) {
    const int wave = blockIdx.x * (blockDim.x >> 5) + (threadIdx.x >> 5);
    const int lane = threadIdx.x & 31;
    const int half = lane >> 4;
    const int rc   = lane & 15;
    const int rowBase = wave * 16;
    if (rowBase >= N_GRAPHS) return;

    v8f acc[8] = {v8f{0}, v8f{0}, v8f{0}, v8f{0}, v8f{0}, v8f{0}, v8f{0}, v8f{0}};
    gemm_core<HID, 8>(G + (size_t)(rowBase + rc) * HID, Wm1, half, rc, acc);

#pragma unroll
    for (int t = 0; t < 8; ++t) {
        const int   col  = t * 16 + rc;
        const float bias = bm1[col];
#pragma unroll
        for (int r = 0; r < 8; ++r) {
            const int m = rowBase + r + 8 * half;
            H[(size_t)m * MLP_HID + col] = fmaxf(acc[t][r] + bias, 0.0f);
        }
    }
}

// ---------------------------------------------------------------------------
// out[g] = dot(H[g, :128], Wm2) + bm2
// ---------------------------------------------------------------------------
__global__ void mlp2_kernel(const float* __restrict__ H,
                            const float* __restrict__ Wm2,
                            const float* __restrict__ bm2,
                            float* __restrict__ out) {
    const int g = blockIdx.x * blockDim.x + threadIdx.x;
    if (g >= N_GRAPHS) return;
    float s = bm2[0];
    const float4* h4 = reinterpret_cast<const float4*>(H + (size_t)g * MLP_HID);
    const float4* w4 = reinterpret_cast<const float4*>(Wm2);
#pragma unroll 8
    for (int i = 0; i < MLP_HID / 4; ++i) {
        const float4 a = h4[i];
        const float4 b = w4[i];
        s += a.x * b.x + a.y * b.y + a.z * b.z + a.w * b.w;
    }
    out[g] = s;
}

extern "C" void kernel_launch(void* const* d_in, const int* in_sizes, int n_in,
                              void* d_out, int out_size, void* d_ws, size_t ws_size,
                              hipStream_t stream) {
    const float* feats = (const float*)d_in[0];
    const int*   src   = (const int*)d_in[1];
    const int*   dst   = (const int*)d_in[2];
    const int*   gid   = (const int*)d_in[3];
    const float* W[3]  = {(const float*)d_in[4],  (const float*)d_in[8],  (const float*)d_in[12]};
    const float* b[3]  = {(const float*)d_in[5],  (const float*)d_in[9],  (const float*)d_in[13]};
    const float* R[3]  = {(const float*)d_in[6],  (const float*)d_in[10], (const float*)d_in[14]};
    const float* rb[3] = {(const float*)d_in[7],  (const float*)d_in[11], (const float*)d_in[15]};
    const float* Wm1 = (const float*)d_in[16];
    const float* bm1 = (const float*)d_in[17];
    const float* Wm2 = (const float*)d_in[18];
    const float* bm2 = (const float*)d_in[19];
    float* out = (float*)d_out;

    // Workspace layout (floats): P0 | P1 | AGG | G | H  (~80 MB total)
    float* ws  = (float*)d_ws;
    const size_t NODE_F = (size_t)N_NODES * HID;           // 6,400,000
    float* P0  = ws;
    float* P1  = ws + NODE_F;
    float* AGG = ws + 2 * NODE_F;
    float* G   = ws + 3 * NODE_F;
    float* H   = G + (size_t)N_GRAPHS * HID;

    const int TPB = 256;
    const int gemmBlocks    = ((N_NODES / 16) + 7) / 8;              // 8 waves/block
    const int scatterBlocks = (int)(((long long)N_EDGES * 16) / TPB);
    const int poolBlocks    = (int)(((long long)N_NODES * 16 + TPB - 1) / TPB);

    // ---- Layer 0: x = feats (K=74), xw->P0, out->P1 ----
    gemm64_kernel<IN_DIM><<<gemmBlocks, TPB, 0, stream>>>(feats, W[0], P0, N_NODES);
    hipMemsetAsync(AGG, 0, NODE_F * sizeof(float), stream);
    scatter_edges_kernel<<<scatterBlocks, TPB, 0, stream>>>(src, dst, P0, AGG);
    combine_kernel<IN_DIM><<<gemmBlocks, TPB, 0, stream>>>(feats, R[0], rb[0], AGG, b[0],
                                                           P1, N_NODES);

    // ---- Layer 1: x = P1 (K=64), xw->P0, out->P0 ----
    gemm64_kernel<HID><<<gemmBlocks, TPB, 0, stream>>>(P1, W[1], P0, N_NODES);
    hipMemsetAsync(AGG, 0, NODE_F * sizeof(float), stream);
    scatter_edges_kernel<<<scatterBlocks, TPB, 0, stream>>>(src, dst, P0, AGG);
    combine_kernel<HID><<<gemmBlocks, TPB, 0, stream>>>(P1, R[1], rb[1], AGG, b[1],
                                                        P0, N_NODES);

    // ---- Layer 2: x = P0 (K=64), xw->P1, out->P1 ----
    gemm64_kernel<HID><<<gemmBlocks, TPB, 0, stream>>>(P0, W[2], P1, N_NODES);
    hipMemsetAsync(AGG, 0, NODE_F * sizeof(float), stream);
    scatter_edges_kernel<<<scatterBlocks, TPB, 0, stream>>>(src, dst, P1, AGG);
    combine_kernel<HID><<<gemmBlocks, TPB, 0, stream>>>(P0, R[2], rb[2], AGG, b[2],
                                                        P1, N_NODES);

    // ---- SumPooling into G ----
    hipMemsetAsync(G, 0, (size_t)N_GRAPHS * HID * sizeof(float), stream);
    pool_kernel<<<poolBlocks, TPB, 0, stream>>>(P1, gid, G);

    // ---- MLP head ----
    const int mlp1Blocks = (N_GRAPHS / 16) / 8;                      // 32 blocks, 256 waves
    mlp1_kernel<<<mlp1Blocks, TPB, 0, stream>>>(G, Wm1, bm1, H);
    mlp2_kernel<<<(N_GRAPHS + TPB - 1) / TPB, TPB, 0, stream>>>(H, Wm2, bm2, out);
}